// PairwiseConv_22290880266595
// MI455X (gfx1250) — compile-verified
//
#include <hip/hip_runtime.h>
#include <hip/hip_bf16.h>

typedef __attribute__((ext_vector_type(2))) float v2f;
typedef __attribute__((ext_vector_type(8))) float v8f;

#define MID 128
#define NROWS 4096          // b*n*h = 2*64*32
#define ROWS 16             // rows per block in kernel B
#define SXS 132             // padded LDS stride for X2 tile (16-row A loads hit distinct banks)
#define SBS 128             // padded LDS stride for basis tile (125 used)
#define SYS 164             // padded LDS stride for Y strip (8*164 % 64 = 32 -> half-waves on distinct banks)
#define OUTDIM 5120         // 32*32*5
#define ROWOUT 25600        // 160*160

// ---------------- Kernel A: x2 = ReLU(LN(ReLU(LN(feat@W1+b1))@W2+b2)) ----------------
__global__ __launch_bounds__(128) void pconv_mlp_kernel(
    const float* __restrict__ feat,
    const float* __restrict__ W1, const float* __restrict__ b1,
    const float* __restrict__ g1, const float* __restrict__ be1,
    const float* __restrict__ W2, const float* __restrict__ b2,
    const float* __restrict__ g2, const float* __restrict__ be2,
    float* __restrict__ X2)
{
    __shared__ float sh[MID];
    __shared__ float red[MID];
    const int r = blockIdx.x;
    const int c = threadIdx.x;

    // layer 1: edge_in == 1 -> outer product
    float x = fmaf(feat[r], W1[c], b1[c]);

    // LayerNorm 1
    red[c] = x; __syncthreads();
    #pragma unroll
    for (int s = 64; s > 0; s >>= 1) { if (c < s) red[c] += red[c + s]; __syncthreads(); }
    const float mu1 = red[0] * (1.0f / 128.0f);
    __syncthreads();
    float dx = x - mu1;
    red[c] = dx * dx; __syncthreads();
    #pragma unroll
    for (int s = 64; s > 0; s >>= 1) { if (c < s) red[c] += red[c + s]; __syncthreads(); }
    const float var1 = red[0] * (1.0f / 128.0f);
    __syncthreads();
    float h = fmaf(dx * __frsqrt_rn(var1 + 1e-5f), g1[c], be1[c]);
    h = fmaxf(h, 0.0f);
    sh[c] = h; __syncthreads();

    // layer 2: 128x128 GEMV per row (W2 is L2-resident, coalesced across threads)
    float acc = b2[c];
    #pragma unroll 8
    for (int k = 0; k < MID; ++k) acc = fmaf(sh[k], W2[k * MID + c], acc);
    __syncthreads();

    // LayerNorm 2
    red[c] = acc; __syncthreads();
    #pragma unroll
    for (int s = 64; s > 0; s >>= 1) { if (c < s) red[c] += red[c + s]; __syncthreads(); }
    const float mu2 = red[0] * (1.0f / 128.0f);
    __syncthreads();
    dx = acc - mu2;
    red[c] = dx * dx; __syncthreads();
    #pragma unroll
    for (int s = 64; s > 0; s >>= 1) { if (c < s) red[c] += red[c + s]; __syncthreads(); }
    const float var2 = red[0] * (1.0f / 128.0f);
    float h2 = fmaf(dx * __frsqrt_rn(var2 + 1e-5f), g2[c], be2[c]);
    X2[r * MID + c] = fmaxf(h2, 0.0f);
}

// ------------- Kernel B: fused y = x2@W3 + b3  (WMMA f32)  +  einsum with basis -------------
// grid = 256 blocks (16 rows each), block = 320 threads = 10 waves.
// WMMA phase : wave w computes Y[0:16, o*160 + 16w : +16] via 32 chained v_wmma_f32_16x16x4_f32.
// Einsum phase: wave w owns output line (row, d); lane = i; basis reads are wave-uniform
//               (LDS broadcast), y reads are stride-5 (bank-conflict-free), stores are
//               5 contiguous floats per lane (640B contiguous per wave).
__global__ __launch_bounds__(320) void pconv_fused_kernel(
    const float* __restrict__ X2,
    const float* __restrict__ W3, const float* __restrict__ b3,
    const float* __restrict__ basis,
    float* __restrict__ out)
{
    __shared__ float sX[ROWS * SXS];    // X2 tile, padded
    __shared__ float sBas[ROWS * SBS];  // basis tile (125 floats/row), padded
    __shared__ float sY[ROWS * SYS];    // one 16x160 Y strip, padded

    const int g = blockIdx.x;
    const int tid = threadIdx.x;

    // stage X2 tile
    for (int idx = tid; idx < ROWS * MID; idx += 320) {
        const int m = idx >> 7, k = idx & 127;
        sX[m * SXS + k] = X2[(g * ROWS + m) * MID + k];
    }
    // stage basis tile: per row 125 floats laid out [d][m][f]
    for (int idx = tid; idx < ROWS * 125; idx += 320) {
        const int m = idx / 125, j = idx - m * 125;
        sBas[m * SBS + j] = basis[(size_t)(g * ROWS + m) * 125 + j];
    }
    __syncthreads();

    const int wave = tid >> 5;
    const int lane = tid & 31;
    const int half = lane >> 4;     // K-half / M-half selector per ISA layout
    const int n16  = lane & 15;     // N (and A's M) index

    const float* srow = &sX[n16 * SXS];           // A row: M = lane&15
    const int wrow = wave / 5;                    // einsum: row offset of this wave
    const int wd   = wave - wrow * 5;             // einsum: fixed d of this wave

    for (int o = 0; o < 32; ++o) {
        // ---- WMMA phase: 16x16 tile, K = 128 in steps of 4 ----
        const int col = o * 160 + wave * 16 + n16;
        v8f acc = {};
        #pragma unroll 4
        for (int kk = 0; kk < 32; ++kk) {
            const int k0 = kk * 4 + half * 2;     // this lane's two K slots in the 4-chunk
            v2f a, bm;
            a.x  = srow[k0];
            a.y  = srow[k0 + 1];
            bm.x = W3[(size_t)k0 * OUTDIM + col];
            bm.y = W3[(size_t)(k0 + 1) * OUTDIM + col];
            acc = __builtin_amdgcn_wmma_f32_16x16x4_f32(
                false, a, false, bm, (short)0, acc, false, false);
        }
        const float bias = b3[col];
        #pragma unroll
        for (int v = 0; v < 8; ++v)               // D: vgpr v, half selects M or M+8
            sY[(v + half * 8) * SYS + wave * 16 + n16] = acc[v] + bias;
        __syncthreads();

        // ---- einsum phase ----
        // out[row, o*5+d, i*5+m] = sum_f Y[row, i*5+f] * B[row, d, m, f]
        // wave -> (row = 2*lp + wrow, d = wd); lane -> i; 8 steps cover 80 lines.
        #pragma unroll 2
        for (int lp = 0; lp < 8; ++lp) {
            const int row = 2 * lp + wrow;
            const float* yrow = &sY[row * SYS + lane * 5];       // stride-5: conflict-free
            const float* brow = &sBas[row * SBS + wd * 25];      // wave-uniform: broadcast

            float yv[5];
            #pragma unroll
            for (int f = 0; f < 5; ++f) yv[f] = yrow[f];

            float* op = &out[(size_t)(g * ROWS + row) * ROWOUT
                             + (size_t)(o * 5 + wd) * 160 + lane * 5];
            #pragma unroll
            for (int m = 0; m < 5; ++m) {
                float s = 0.0f;
                #pragma unroll
                for (int f = 0; f < 5; ++f) s = fmaf(yv[f], brow[m * 5 + f], s);
                op[m] = s;
            }
        }
        __syncthreads();   // sY reused next o
    }
}

extern "C" void kernel_launch(void* const* d_in, const int* in_sizes, int n_in,
                              void* d_out, int out_size, void* d_ws, size_t ws_size,
                              hipStream_t stream) {
    const float* feat  = (const float*)d_in[0];
    const float* basis = (const float*)d_in[1];
    const float* W1    = (const float*)d_in[2];
    const float* b1    = (const float*)d_in[3];
    const float* g1    = (const float*)d_in[4];
    const float* beta1 = (const float*)d_in[5];
    const float* W2    = (const float*)d_in[6];
    const float* b2    = (const float*)d_in[7];
    const float* g2    = (const float*)d_in[8];
    const float* beta2 = (const float*)d_in[9];
    const float* W3    = (const float*)d_in[10];
    const float* b3    = (const float*)d_in[11];
    float* out = (float*)d_out;
    float* X2  = (float*)d_ws;   // 4096*128*4 = 2 MB scratch

    pconv_mlp_kernel<<<NROWS, 128, 0, stream>>>(feat, W1, b1, g1, beta1,
                                                W2, b2, g2, beta2, X2);
    pconv_fused_kernel<<<NROWS / ROWS, 320, 0, stream>>>(X2, W3, b3, basis, out);
}